// ModelA_46394236732084
// MI455X (gfx1250) — compile-verified
//
#include <hip/hip_runtime.h>

typedef float v2f __attribute__((ext_vector_type(2)));
typedef float v8f __attribute__((ext_vector_type(8)));
typedef int   v4i __attribute__((ext_vector_type(4)));

#define N_NODES 50000
#define N_EDGES 800000
#define NFEAT   384

#if __has_builtin(__builtin_amdgcn_global_load_async_to_lds_b128)
#define HAVE_ASYNC_COPY 1
typedef __attribute__((address_space(1))) v4i* gp4_t;   // global v4i*
typedef __attribute__((address_space(3))) v4i* lp4_t;   // LDS v4i*
#endif

__device__ __forceinline__ float lrelu(float x) { return x >= 0.f ? x : 0.01f * x; }

// 16-byte global -> LDS copy (async on CDNA5 when available)
__device__ __forceinline__ void copy16(const float* src, float* lds_dst) {
#ifdef HAVE_ASYNC_COPY
  __builtin_amdgcn_global_load_async_to_lds_b128((gp4_t)src, (lp4_t)lds_dst, 0, 0);
#else
  *(float4*)lds_dst = *(const float4*)src;
#endif
}

__device__ __forceinline__ void async_wait_all() {
#ifdef HAVE_ASYNC_COPY
#if __has_builtin(__builtin_amdgcn_s_wait_asynccnt)
  __builtin_amdgcn_s_wait_asynccnt(0);
#else
  asm volatile("s_wait_asynccnt 0x0" ::: "memory");
#endif
#endif
}

// ---------------------------------------------------------------- degree prep
__global__ void k_fill(float* __restrict__ p, float v, int n) {
  int i = blockIdx.x * blockDim.x + threadIdx.x;
  if (i < n) p[i] = v;
}

__global__ void k_degree(const int* __restrict__ ei, float* __restrict__ deg, int E) {
  int e = blockIdx.x * blockDim.x + threadIdx.x;
  if (e < E) atomicAdd(&deg[ei[E + e]], 1.0f);
}

__global__ void k_rsqrt(float* __restrict__ p, int n) {
  int i = blockIdx.x * blockDim.x + threadIdx.x;
  if (i < n) p[i] = rsqrtf(p[i]);
}

// ---------------------------------------------------------------- fp32 WMMA GEMM
// Out[M,F] = act(A[M,K]) @ W[K,F]; MODE==1: Out = relu(.. + bias)
// Block = 128 threads (4 waves). Block tile 64x64; each wave: 16 rows x 64 cols
// (4 v8f accumulators). K chunk = 32, double-buffered async LDS staging.
// ACT/MODE are compile-time so no predicated dead paths remain in the hot loop.
template <int ACT, int MODE>
__global__ __launch_bounds__(128)
void k_gemm_wmma(const float* __restrict__ A, const float* __restrict__ W,
                 const float* __restrict__ bias, float* __restrict__ Out,
                 int M, int K, int F) {
  __shared__ float As[2][64 * 35];   // stride 35: half-waves hit disjoint banks
  __shared__ float Bs[2][32 * 72];   // stride 72: +2 rows => +16 banks (disjoint)

  const int tid  = threadIdx.x;
  const int wave = tid >> 5;
  const int lane = tid & 31;
  const int hi   = lane >> 4;
  const int lo   = lane & 15;
  const int row0 = blockIdx.x * 64;
  const int col0 = blockIdx.y * 64;

  v8f acc[4] = {};

  auto stage = [&](int k0, int buf) {
    // A tile: 64 rows x 32 cols, 512 x b128, 4 per thread
    #pragma unroll
    for (int j = 0; j < 4; ++j) {
      int i  = tid + j * 128;
      int r  = i >> 3;
      int c4 = (i & 7) << 2;
      int gr = row0 + r;
      if (gr < M)   // skipped rows leave garbage; only fully-invalid subtiles see it
        copy16(A + (size_t)gr * K + k0 + c4, &As[buf][r * 35 + c4]);
    }
    // B tile: 32 rows x 64 cols, 512 x b128, 4 per thread
    #pragma unroll
    for (int j = 0; j < 4; ++j) {
      int i  = tid + j * 128;
      int r  = i >> 4;
      int c4 = (i & 15) << 2;
      copy16(W + (size_t)(k0 + r) * F + col0 + c4, &Bs[buf][r * 72 + c4]);
    }
  };

  stage(0, 0);
  const int nk = K >> 5;
  for (int kc = 0; kc < nk; ++kc) {
    async_wait_all();      // own copies for buffer kc&1 have landed
    __syncthreads();       // everyone's copies visible; prior compute retired
    if (kc + 1 < nk) stage((kc + 1) << 5, (kc + 1) & 1);

    const float* as = &As[kc & 1][0];
    const float* bs = &Bs[kc & 1][0];
    const int mrow = wave * 16 + lo;
    #pragma unroll
    for (int kk = 0; kk < 32; kk += 4) {
      // A 16x4 f32 frag: lane -> row = lane%16, K = kk + 2*(lane/16) + {0,1}
      int kb = kk + 2 * hi;
      v2f a;
      a.x = as[mrow * 35 + kb];
      a.y = as[mrow * 35 + kb + 1];
      if (ACT) { a.x = lrelu(a.x); a.y = lrelu(a.y); }   // fused leaky_relu
      #pragma unroll
      for (int t = 0; t < 4; ++t) {
        v2f b;
        b.x = bs[kb * 72 + t * 16 + lo];
        b.y = bs[(kb + 1) * 72 + t * 16 + lo];
        acc[t] = __builtin_amdgcn_wmma_f32_16x16x4_f32(
            false, a, false, b, (short)0, acc[t], false, false);
      }
    }
  }

  // C/D layout: VGPR j -> M = j + 8*(lane/16), N = lane%16
  #pragma unroll
  for (int t = 0; t < 4; ++t) {
    #pragma unroll
    for (int j = 0; j < 8; ++j) {
      int gr = row0 + wave * 16 + hi * 8 + j;
      int gc = col0 + t * 16 + lo;
      if (gr < M) {
        float v = acc[t][j];
        if (MODE == 1) { v += bias[gc]; v = fmaxf(v, 0.f); }
        Out[(size_t)gr * F + gc] = v;
      }
    }
  }
}

// ---------------------------------------------------------------- GCN aggregate
// agg[i,:] = bias + dinv[i]^2 * hw[i,:]   (self-loop term folded in)
__global__ void k_init_agg(const float* __restrict__ hw, const float* __restrict__ bias,
                           const float* __restrict__ dinv, float* __restrict__ agg,
                           int Nn, int F) {
  int chunks = F >> 2;
  int idx = blockIdx.x * blockDim.x + threadIdx.x;
  if (idx >= Nn * chunks) return;
  int i = idx / chunks;
  int c = (idx - i * chunks) << 2;
  float di = dinv[i];
  float w  = di * di;
  float4 v = *(const float4*)(hw + (size_t)i * F + c);
  float4 b = *(const float4*)(bias + c);
  float4 o = make_float4(b.x + w * v.x, b.y + w * v.y, b.z + w * v.z, b.w + w * v.w);
  *(float4*)(agg + (size_t)i * F + c) = o;
}

// agg[dst,:] += dinv[src]*dinv[dst] * hw[src,:]  over all edges
__global__ void k_scatter(const float* __restrict__ hw, float* __restrict__ agg,
                          const int* __restrict__ ei, const float* __restrict__ dinv,
                          int E, int F) {
  int chunks = F >> 2;                 // 32/64/96 -> edge uniform per wave
  long long idx = (long long)blockIdx.x * blockDim.x + threadIdx.x;
  if (idx >= (long long)E * chunks) return;
  int e = (int)(idx / chunks);
  int c = (int)(idx - (long long)e * chunks) << 2;
  int s = ei[e];
  int d = ei[E + e];
  float w = dinv[s] * dinv[d];
  float4 v = *(const float4*)(hw + (size_t)s * F + c);
  float* p = agg + (size_t)d * F + c;
  atomicAdd(p + 0, w * v.x);
  atomicAdd(p + 1, w * v.y);
  atomicAdd(p + 2, w * v.z);
  atomicAdd(p + 3, w * v.w);
}

// ---------------------------------------------------------------- host wiring
template <int ACT>
static void run_layer(const float* in, const float* Wm, const float* bv,
                      float* hw, float* agg, const float* dinv, const int* ei,
                      int K, int F, hipStream_t stream) {
  dim3 g((N_NODES + 63) / 64, F / 64);
  k_gemm_wmma<ACT, 0><<<g, 128, 0, stream>>>(in, Wm, nullptr, hw, N_NODES, K, F);
  int t1 = N_NODES * (F >> 2);
  k_init_agg<<<(t1 + 255) / 256, 256, 0, stream>>>(hw, bv, dinv, agg, N_NODES, F);
  long long t2 = (long long)N_EDGES * (F >> 2);
  k_scatter<<<(unsigned)((t2 + 255) / 256), 256, 0, stream>>>(hw, agg, ei, dinv, N_EDGES, F);
}

extern "C" void kernel_launch(void* const* d_in, const int* in_sizes, int n_in,
                              void* d_out, int out_size, void* d_ws, size_t ws_size,
                              hipStream_t stream) {
  const float* x  = (const float*)d_in[0];
  const int*   ei = (const int*)d_in[1];
  const float* W1 = (const float*)d_in[2];
  const float* b1 = (const float*)d_in[3];
  const float* W2 = (const float*)d_in[4];
  const float* b2 = (const float*)d_in[5];
  const float* W3 = (const float*)d_in[6];
  const float* b3 = (const float*)d_in[7];
  const float* W4 = (const float*)d_in[8];
  const float* b4 = (const float*)d_in[9];
  const float* Wl = (const float*)d_in[10];
  const float* bl = (const float*)d_in[11];

  float* dinv = (float*)d_ws;                       // N floats (deg -> rsqrt in place)
  float* bufA = dinv + 50048;                       // N x 384 (hw)
  float* bufB = bufA + (size_t)N_NODES * NFEAT;     // N x 384 (agg / next input)

  // symmetric-norm degrees (self-loop => init 1.0)
  k_fill<<<(N_NODES + 255) / 256, 256, 0, stream>>>(dinv, 1.0f, N_NODES);
  k_degree<<<(N_EDGES + 255) / 256, 256, 0, stream>>>(ei, dinv, N_EDGES);
  k_rsqrt<<<(N_NODES + 255) / 256, 256, 0, stream>>>(dinv, N_NODES);

  // 4 GCN layers (leaky_relu fused into next GEMM's A-fragment read)
  run_layer<0>(x,    W1, b1, bufA, bufB, dinv, ei, NFEAT, 128,   stream);
  run_layer<1>(bufB, W2, b2, bufA, bufB, dinv, ei, 128,   NFEAT, stream);
  run_layer<1>(bufB, W3, b3, bufA, bufB, dinv, ei, NFEAT, 256,   stream);
  run_layer<1>(bufB, W4, b4, bufA, bufB, dinv, ei, 256,   NFEAT, stream);

  // head: out = relu(leaky_relu(agg4) @ Wl + bl)
  dim3 g((N_NODES + 63) / 64, 128 / 64);
  k_gemm_wmma<1, 1><<<g, 128, 0, stream>>>(bufB, Wl, bl, (float*)d_out,
                                           N_NODES, NFEAT, 128);
}